// Tensor_CSPNet_23484881174580
// MI455X (gfx1250) — compile-verified
//
#include <hip/hip_runtime.h>
#include <hip/hip_bf16.h>

// ---------------------------------------------------------------------------
// SPDNet forward for MI455X (gfx1250, wave32).
// All spectral ops (ReEig / LogEig / BN sqrt) are reformulated as matmul-only
// Newton-Schulz iterations so the entire pipeline runs on V_WMMA_F32_16X16X4_F32.
// One wave per (batch, channel) SPD matrix; matrices live in LDS (stride 33,
// bank-conflict-free fragment loads); 2x2 tiles of 16x16 f32 WMMA fragments.
// ---------------------------------------------------------------------------

typedef __attribute__((ext_vector_type(2))) float v2f;
typedef __attribute__((ext_vector_type(8))) float v8f;

#define LSTR 33
#define BUFSZ (32 * LSTR)       // 1056 floats per 32x32 (padded) matrix buffer
#define WAVES_PER_BLOCK 2
#define EPS 1e-4f
#define NS_SIGN_ITERS 12        // ReEig |A| sign iteration
#define SQRT_LEVELS 6           // LogEig inverse scaling-and-squaring depth
#define SQRT_NS_ITERS 7         // coupled NS sqrt inner iterations
#define LOG_SERIES 7            // Mercator series order

// ---- wave-local LDS producer/consumer ordering (single wave per buffer set)
__device__ __forceinline__ void wsync() {
  __builtin_amdgcn_fence(__ATOMIC_RELEASE, "workgroup");
  __builtin_amdgcn_wave_barrier();
  __builtin_amdgcn_fence(__ATOMIC_ACQUIRE, "workgroup");
}

// ---- WMMA fragment loads (ISA 7.12.2 layouts, stride 33 => conflict-free)
// A 16x4: lane l -> row m0+(l&15); VGPR0/1 hold K = k0+2*(l>>4) and +1
__device__ __forceinline__ v2f ldA(const float* M, int m0, int k0, int lane) {
  int r  = m0 + (lane & 15);
  int kb = k0 + ((lane >> 4) << 1);
  const float* p = M + r * LSTR + kb;
  v2f a; a.x = p[0]; a.y = p[1]; return a;
}
// B 4x16: lane l -> col n0+(l&15); VGPR0/1 hold K = k0+2*(l>>4) and +1
__device__ __forceinline__ v2f ldB(const float* M, int k0, int n0, int lane) {
  int n  = n0 + (lane & 15);
  int kb = k0 + ((lane >> 4) << 1);
  v2f b; b.x = M[kb * LSTR + n]; b.y = M[(kb + 1) * LSTR + n]; return b;
}
// B = M^T : B[k][n] = M[n][k]
__device__ __forceinline__ v2f ldBT(const float* M, int k0, int n0, int lane) {
  int n  = n0 + (lane & 15);
  int kb = k0 + ((lane >> 4) << 1);
  const float* p = M + n * LSTR + kb;
  v2f b; b.x = p[0]; b.y = p[1]; return b;
}

// D(32x32) = A(32xK) * B(Kx32), K multiple of 4, accumulators in registers
template <bool BT>
__device__ __forceinline__ void mm32(const float* A, const float* B, int K,
                                     int lane, v8f (&acc)[2][2]) {
#pragma unroll
  for (int t = 0; t < 2; ++t)
#pragma unroll
    for (int u = 0; u < 2; ++u) acc[t][u] = (v8f){0, 0, 0, 0, 0, 0, 0, 0};
#pragma unroll
  for (int k = 0; k < K; k += 4) {
    v2f a0 = ldA(A, 0, k, lane);
    v2f a1 = ldA(A, 16, k, lane);
    v2f b0, b1;
    if (BT) { b0 = ldBT(B, k, 0, lane); b1 = ldBT(B, k, 16, lane); }
    else    { b0 = ldB(B, k, 0, lane);  b1 = ldB(B, k, 16, lane);  }
    acc[0][0] = __builtin_amdgcn_wmma_f32_16x16x4_f32(false, a0, false, b0, (short)0, acc[0][0], false, false);
    acc[0][1] = __builtin_amdgcn_wmma_f32_16x16x4_f32(false, a0, false, b1, (short)0, acc[0][1], false, false);
    acc[1][0] = __builtin_amdgcn_wmma_f32_16x16x4_f32(false, a1, false, b0, (short)0, acc[1][0], false, false);
    acc[1][1] = __builtin_amdgcn_wmma_f32_16x16x4_f32(false, a1, false, b1, (short)0, acc[1][1], false, false);
  }
}

// Store C fragments: D = sAcc*acc + sAdd*addSrc + diagAdd*I (+ jitter diag)
__device__ __forceinline__ void storeC(float* D, const float* addSrc, int lane,
                                       const v8f (&acc)[2][2], float sAcc,
                                       float sAdd, float diagAdd, bool jitter) {
  int n0 = lane & 15;
  int rb = 8 * (lane >> 4);
#pragma unroll
  for (int t = 0; t < 2; ++t)
#pragma unroll
    for (int u = 0; u < 2; ++u)
#pragma unroll
      for (int r = 0; r < 8; ++r) {
        int row = 16 * t + rb + r;
        int col = 16 * u + n0;
        float v = sAcc * acc[t][u][r];
        if (addSrc) v += sAdd * addSrc[row * LSTR + col];
        if (row == col) {
          v += diagAdd;
          if (jitter) v += 1.0e-6f + (1.0e-6f / 31.0f) * (float)row;
        }
        D[row * LSTR + col] = v;
      }
}

__device__ __forceinline__ float wredsum(float s) {
#pragma unroll
  for (int off = 16; off > 0; off >>= 1) s += __shfl_xor(s, off, 32);
  return s;
}
__device__ __forceinline__ float frob(const float* M, int lane) {
  float s = 0.f;
#pragma unroll
  for (int r = 0; r < 32; ++r) { float v = M[r * LSTR + lane]; s += v * v; }
  return sqrtf(wredsum(s));
}
__device__ __forceinline__ float traceM(const float* M, int lane) {
  return wredsum(M[lane * LSTR + lane]);
}

// ReEig(X) = 0.5*(X + eps I + |X - eps I|); |A| = c0 * Z0 @ sign(Z0) via NS.
// Input and output in b0. b1..b3 scratch.
__device__ __forceinline__ void reig32(float* b0, float* b1, float* b2,
                                       float* b3, int lane) {
  v8f acc[2][2];
  b0[lane * LSTR + lane] -= EPS;   // A0 = X - eps I
  wsync();
  float c0 = frob(b0, lane) + 1e-30f;
  float inv = 1.0f / c0;
#pragma unroll
  for (int r = 0; r < 32; ++r) b0[r * LSTR + lane] *= inv;  // Z0 = A0/c0
  wsync();
  float* S = b1; float* Q = b2; float* T4 = b3;
  for (int i = lane; i < BUFSZ; i += 32) S[i] = b0[i];
  wsync();
#pragma clang loop unroll(disable)
  for (int it = 0; it < NS_SIGN_ITERS; ++it) {
    mm32<false>(S, S, 32, lane, acc);                     // P = S*S
    storeC(Q, nullptr, lane, acc, -1.f, 0.f, 3.f, false); // Q = 3I - P
    wsync();
    mm32<false>(S, Q, 32, lane, acc);                     // S' = 0.5 S Q
    storeC(T4, nullptr, lane, acc, 0.5f, 0.f, 0.f, false);
    wsync();
    float* t = S; S = T4; T4 = t;
  }
  // R = 0.5*c0*(Z0 + Z0@S) + eps I
  mm32<false>(b0, S, 32, lane, acc);
  storeC(Q, b0, lane, acc, 0.5f * c0, 0.5f * c0, EPS, false);
  wsync();
  for (int i = lane; i < BUFSZ; i += 32) b0[i] = Q[i];
  wsync();
}

// log(X) = log(tr) I + 2^m * MercatorSeries(X_hat^(1/2^m) - I).
// Input in b0; returns pointer to result buffer.
__device__ __forceinline__ float* logeig32(float* b0, float* b1, float* b2,
                                           float* b3, int lane) {
  v8f acc[2][2];
  float* Y = b0; float* Z = b1; float* Q = b2; float* W = b3;
  float s = traceM(Y, lane);                    // >= lambda_max (SPD)
  float inv = 1.0f / s;
#pragma unroll
  for (int r = 0; r < 32; ++r) Y[r * LSTR + lane] *= inv;
  wsync();
#pragma clang loop unroll(disable)
  for (int lev = 0; lev < SQRT_LEVELS; ++lev) {  // Y <- sqrt(Y) (coupled NS)
    for (int i = lane; i < BUFSZ; i += 32) Z[i] = 0.f;
    wsync();
    Z[lane * LSTR + lane] = 1.0f;
    wsync();
#pragma clang loop unroll(disable)
    for (int it = 0; it < SQRT_NS_ITERS; ++it) {
      mm32<false>(Z, Y, 32, lane, acc);                     // P = Z Y
      storeC(Q, nullptr, lane, acc, -1.f, 0.f, 3.f, false); // Q = 3I - P
      wsync();
      mm32<false>(Y, Q, 32, lane, acc);                     // Ynew = 0.5 Y Q
      storeC(W, nullptr, lane, acc, 0.5f, 0.f, 0.f, false);
      wsync();
      mm32<false>(Q, Z, 32, lane, acc);                     // Znew = 0.5 Q Z
      storeC(Y, nullptr, lane, acc, 0.5f, 0.f, 0.f, false);
      wsync();
      float* oy = Y; float* oz = Z;
      Y = W; Z = oy; W = oz;   // rotate: Y<-Ynew, Z<-Znew(old Y slot), free<-old Z
    }
  }
  // E = Y - I
  for (int i = lane; i < BUFSZ; i += 32) Z[i] = Y[i];
  wsync();
  Z[lane * LSTR + lane] -= 1.0f;
  wsync();
  float cJ = ((LOG_SERIES & 1) ? 1.f : -1.f) / (float)LOG_SERIES;
  for (int i = lane; i < BUFSZ; i += 32) Q[i] = cJ * Z[i];   // G = cJ*E
  wsync();
#pragma clang loop unroll(disable)
  for (int j = LOG_SERIES - 1; j >= 1; --j) {   // Horner: G = E*G + cj*E
    float cj = ((j & 1) ? 1.f : -1.f) / (float)j;
    mm32<false>(Z, Q, 32, lane, acc);
    storeC(W, Z, lane, acc, 1.f, cj, 0.f, false);
    wsync();
    float* t = Q; Q = W; W = t;
  }
  float lgs = logf(s);
  float p2 = (float)(1 << SQRT_LEVELS);
  for (int i = lane; i < BUFSZ; i += 32) {
    int r = i / LSTR, cc = i % LSTR;
    float v = p2 * Q[i];
    if (r == cc) v += lgs;
    W[i] = v;
  }
  wsync();
  return W;
}

// ---------------------------------------------------------------------------
// Kernel A: T = G^{1/2} M^{-1/2} via coupled NS sqrt (gives A^{1/2} and
// A^{-1/2} simultaneously). One wave. Stores dense 32x32 T (I-padded) in ws.
// ---------------------------------------------------------------------------
__global__ __launch_bounds__(32) void spdnet_bn_prep(
    const float* __restrict__ bn_mean, const float* __restrict__ bn_weight,
    float* __restrict__ ws) {
  __shared__ float smem[4 * BUFSZ];
  int lane = threadIdx.x & 31;
  v8f acc[2][2];
#pragma clang loop unroll(disable)
  for (int pass = 0; pass < 2; ++pass) {
    const float* src = (pass == 0) ? bn_weight : bn_mean;
    for (int i = lane; i < 4 * BUFSZ; i += 32) smem[i] = 0.f;
    wsync();
    for (int i = lane; i < 484; i += 32)
      smem[(i / 22) * LSTR + (i % 22)] = src[i];
    wsync();
    if (lane >= 22) smem[lane * LSTR + lane] = 1.0f;  // unit padding block
    wsync();
    float* Y = smem; float* Z = smem + BUFSZ;
    float* Q = smem + 2 * BUFSZ; float* W = smem + 3 * BUFSZ;
    float s = traceM(Y, lane);
    float inv = 1.0f / s;
#pragma unroll
    for (int r = 0; r < 32; ++r) Y[r * LSTR + lane] *= inv;
    wsync();
    Z[lane * LSTR + lane] = 1.0f;
    wsync();
#pragma clang loop unroll(disable)
    for (int it = 0; it < 14; ++it) {
      mm32<false>(Z, Y, 32, lane, acc);
      storeC(Q, nullptr, lane, acc, -1.f, 0.f, 3.f, false);
      wsync();
      mm32<false>(Y, Q, 32, lane, acc);
      storeC(W, nullptr, lane, acc, 0.5f, 0.f, 0.f, false);
      wsync();
      mm32<false>(Q, Z, 32, lane, acc);
      storeC(Y, nullptr, lane, acc, 0.5f, 0.f, 0.f, false);
      wsync();
      float* oy = Y; float* oz = Z;
      Y = W; Z = oy; W = oz;
    }
    if (pass == 0) {
      float f = sqrtf(s);  // G^{1/2} = sqrt(s)*Y -> ws[1024..2047]
      for (int i = lane; i < 1024; i += 32)
        ws[1024 + i] = f * Y[(i >> 5) * LSTR + (i & 31)];
    } else {
      float f = 1.0f / sqrtf(s);  // M^{-1/2} = Z/sqrt(s)
#pragma unroll
      for (int r = 0; r < 32; ++r) Z[r * LSTR + lane] *= f;
      wsync();
      for (int i = lane; i < 1024; i += 32)  // reload G^{1/2} into Y
        Y[(i >> 5) * LSTR + (i & 31)] = ws[1024 + i];
      wsync();
      mm32<false>(Y, Z, 32, lane, acc);      // T = G^{1/2} @ M^{-1/2}
      storeC(Q, nullptr, lane, acc, 1.f, 0.f, 0.f, false);
      wsync();
      for (int i = lane; i < 1024; i += 32)
        ws[i] = Q[(i >> 5) * LSTR + (i & 31)];
    }
  }
}

// ---------------------------------------------------------------------------
// Kernel B: per-(b,c) full SPD pipeline, one wave per matrix.
// ---------------------------------------------------------------------------
__global__ __launch_bounds__(WAVES_PER_BLOCK * 32) void spdnet_main(
    const float* __restrict__ x, const float* __restrict__ W1,
    const float* __restrict__ W2, const float* __restrict__ Tws,
    const float* __restrict__ conv_w, float* __restrict__ ypart) {
  __shared__ float smem[WAVES_PER_BLOCK * 4 * BUFSZ];
  int lane = threadIdx.x & 31;
  int wid  = threadIdx.x >> 5;
  int midx = blockIdx.x * WAVES_PER_BLOCK + wid;  // 0..55295
  int b = midx / 27, c = midx % 27;
  float* b0 = smem + wid * 4 * BUFSZ;
  float* b1 = b0 + BUFSZ; float* b2 = b0 + 2 * BUFSZ; float* b3 = b0 + 3 * BUFSZ;
  v8f acc[2][2];

  for (int i = lane; i < 4 * BUFSZ; i += 32) b0[i] = 0.f;
  wsync();
  // load X (22x22, zero padded) and W1_c (32x22)
  const float* Xg = x + (size_t)midx * 484;
  for (int i = lane; i < 484; i += 32) b0[(i / 22) * LSTR + (i % 22)] = Xg[i];
  const float* W1g = W1 + (size_t)c * 704;
  for (int i = lane; i < 704; i += 32) b1[(i / 22) * LSTR + (i % 22)] = W1g[i];
  wsync();
  // BiMap1: Y = W1 X W1^T + jitter diag
  mm32<false>(b1, b0, 24, lane, acc);            // tmp = W1 @ X
  storeC(b2, nullptr, lane, acc, 1.f, 0.f, 0.f, false);
  wsync();
  mm32<true>(b2, b1, 24, lane, acc);             // Y = tmp @ W1^T (+jitter)
  storeC(b0, nullptr, lane, acc, 1.f, 0.f, 0.f, true);
  wsync();
  reig32(b0, b1, b2, b3, lane);                  // ReEig (32x32)
  // BiMap2: Z = W2 R W2^T  (22x22, zero padded)
  for (int i = lane; i < BUFSZ; i += 32) b1[i] = 0.f;
  wsync();
  const float* W2g = W2 + (size_t)c * 704;
  for (int i = lane; i < 704; i += 32) b1[(i / 32) * LSTR + (i % 32)] = W2g[i];
  wsync();
  mm32<false>(b1, b0, 32, lane, acc);            // tmp = W2 @ R
  storeC(b2, nullptr, lane, acc, 1.f, 0.f, 0.f, false);
  wsync();
  mm32<true>(b2, b1, 32, lane, acc);             // Z = tmp @ W2^T
  storeC(b3, nullptr, lane, acc, 1.f, 0.f, 0.f, false);
  wsync();
  // BatchNormSPD: Z <- T Z T^T  (T is I-padded 32x32 from prep kernel)
  for (int i = lane; i < 1024; i += 32)
    b0[(i >> 5) * LSTR + (i & 31)] = Tws[i];
  wsync();
  mm32<false>(b0, b3, 32, lane, acc);            // T @ Z
  storeC(b1, nullptr, lane, acc, 1.f, 0.f, 0.f, false);
  wsync();
  mm32<true>(b1, b0, 32, lane, acc);             // @ T^T
  storeC(b2, nullptr, lane, acc, 1.f, 0.f, 0.f, false);
  wsync();
  // move to b0, set padding diag to 1 (keeps padding spectrum at {1})
  for (int i = lane; i < BUFSZ; i += 32) b0[i] = b2[i];
  wsync();
  if (lane >= 22) b0[lane * LSTR + lane] = 1.0f;
  wsync();
  reig32(b0, b1, b2, b3, lane);                  // ReEig (22x22 block)
  float* LG = logeig32(b0, b1, b2, b3, lane);    // LogEig

  // Folded conv head: y[b,o] += sum_ij LG[i,j] * conv_w[o, c/9, (c%9)*484 + ij]
  int h = c / 9, bandi = c % 9;
  const float* cwb = conv_w + (size_t)h * 4356 + (size_t)bandi * 484;
  float* yb = ypart + (size_t)b * 48;
#pragma clang loop unroll(disable)
  for (int o = 0; o < 48; ++o) {
    const float* cw = cwb + (size_t)o * 13068;   // 3*4356
    float sum = 0.f;
    for (int e = lane; e < 484; e += 32)
      sum += LG[(e / 22) * LSTR + (e % 22)] * cw[e];
    sum = wredsum(sum);
    if (lane == 0) atomicAdd(&yb[o], sum);
  }
}

// ---------------------------------------------------------------------------
// Kernel C: out = (y + conv_b) @ fc_w^T + fc_b
// ---------------------------------------------------------------------------
__global__ void spdnet_head(const float* __restrict__ ypart,
                            const float* __restrict__ conv_b,
                            const float* __restrict__ fc_w,
                            const float* __restrict__ fc_b,
                            float* __restrict__ out) {
  int bidx = blockIdx.x * blockDim.x + threadIdx.x;
  if (bidx >= 2048) return;
  const float* yb = ypart + (size_t)bidx * 48;
  float s0 = fc_b[0], s1 = fc_b[1], s2 = fc_b[2], s3 = fc_b[3];
#pragma unroll
  for (int o = 0; o < 48; ++o) {
    float v = yb[o] + conv_b[o];
    s0 += v * fc_w[0 * 48 + o];
    s1 += v * fc_w[1 * 48 + o];
    s2 += v * fc_w[2 * 48 + o];
    s3 += v * fc_w[3 * 48 + o];
  }
  float* ob = out + (size_t)bidx * 4;
  ob[0] = s0; ob[1] = s1; ob[2] = s2; ob[3] = s3;
}

extern "C" void kernel_launch(void* const* d_in, const int* in_sizes, int n_in,
                              void* d_out, int out_size, void* d_ws,
                              size_t ws_size, hipStream_t stream) {
  (void)in_sizes; (void)n_in; (void)out_size; (void)ws_size;
  const float* x         = (const float*)d_in[0];
  const float* W1        = (const float*)d_in[1];
  const float* W2        = (const float*)d_in[2];
  const float* bn_mean   = (const float*)d_in[3];
  const float* bn_weight = (const float*)d_in[4];
  const float* conv_w    = (const float*)d_in[5];
  const float* conv_b    = (const float*)d_in[6];
  const float* fc_w      = (const float*)d_in[7];
  const float* fc_b      = (const float*)d_in[8];

  float* ws    = (float*)d_ws;
  float* Tbuf  = ws;            // [0..1023] T (32x32), [1024..2047] G^{1/2} scratch
  float* ypart = ws + 2048;     // 2048*48 partial conv accumulators

  hipMemsetAsync(ypart, 0, (size_t)2048 * 48 * sizeof(float), stream);
  spdnet_bn_prep<<<1, 32, 0, stream>>>(bn_mean, bn_weight, ws);
  const int nmat = 2048 * 27;
  spdnet_main<<<nmat / WAVES_PER_BLOCK, WAVES_PER_BLOCK * 32, 0, stream>>>(
      x, W1, W2, Tbuf, conv_w, ypart);
  spdnet_head<<<(2048 + 255) / 256, 256, 0, stream>>>(ypart, conv_b, fc_w,
                                                      fc_b, (float*)d_out);
}